// DenseOSTL_30940944400972
// MI455X (gfx1250) — compile-verified
//
#include <hip/hip_runtime.h>

typedef __attribute__((ext_vector_type(2))) float v2f;
typedef __attribute__((ext_vector_type(4))) float f32x4;
typedef __attribute__((ext_vector_type(8))) float v8f;

#define BETA 0.9f

// ---------------------------------------------------------------------------
// Kernel 1: I = x@W + b via V_WMMA_F32_16X16X4_F32, fused LIF epilogue.
// One wave per 16x16 tile of the [64,1024] output. 256 tiles = 32 blocks x 8 waves.
// Writes spk, u_new, E_b_new to d_out and one_m to workspace.
// ---------------------------------------------------------------------------
__global__ __launch_bounds__(256) void dense_lif_wmma(
    const float* __restrict__ x,     // [64,1024]
    const float* __restrict__ W,     // [1024,1024]
    const float* __restrict__ bias,  // [1024]
    const float* __restrict__ u0,    // [64,1024]
    const float* __restrict__ E_b,   // [64,1024]
    float* __restrict__ out_spk,     // [64,1024]
    float* __restrict__ out_u,       // [64,1024]
    float* __restrict__ out_Eb,      // [64,1024]
    float* __restrict__ ws_one_m)    // [64,1024]
{
    const int lane = threadIdx.x & 31;
    const int wave = threadIdx.x >> 5;
    const int tile = blockIdx.x * 8 + wave;   // 0..255
    const int tm   = tile >> 6;               // 0..3   (row tile, 64/16)
    const int tn   = tile & 63;               // 0..63  (col tile, 1024/16)
    const int half = lane >> 4;               // 0 or 1
    const int lr   = lane & 15;

    const int Din = 1024, Dout = 1024;
    const float* xrow = x + (tm * 16 + lr) * Din;   // A: row M=lr of tile
    const float* Wcol = W + (tn * 16 + lr);         // B: col N=lr of tile

    v8f c = {};
    for (int k = 0; k < Din; k += 4) {
        // A 16x4 f32: lane L holds K = (L>=16?2:0)+r in vgpr r
        v2f a, bm;
        a.x  = xrow[k + half * 2 + 0];
        a.y  = xrow[k + half * 2 + 1];
        // B 4x16 f32: lane L holds column N=L%16, K = (L>=16?2:0)+r in vgpr r
        bm.x = Wcol[(size_t)(k + half * 2 + 0) * Dout];
        bm.y = Wcol[(size_t)(k + half * 2 + 1) * Dout];
        c = __builtin_amdgcn_wmma_f32_16x16x4_f32(
                /*neg_a=*/false, a, /*neg_b=*/false, bm,
                /*c_mod=*/(short)0, c, /*reuse_a=*/false, /*reuse_b=*/false);
    }

    // Epilogue: C/D layout -> element (m = tm*16 + half*8 + r, n = tn*16 + lr)
    const int   n  = tn * 16 + lr;
    const float bn = bias[n];
#pragma unroll
    for (int r = 0; r < 8; ++r) {
        const int m   = tm * 16 + half * 8 + r;
        const int idx = m * Dout + n;
        const float I     = c[r] + bn;
        const float v     = fmaf(BETA, u0[idx], I);
        const float vt    = v - 1.0f;                 // THRESH = 1
        const float spk   = (vt > 0.0f) ? 1.0f : 0.0f;
        const float d     = 1.0f + fabsf(vt);
        const float sg    = 1.0f / (d * d);           // fast-sigmoid surrogate
        const float u_new = v - spk;                  // soft reset (THRESH=1)
        const float one_m = 1.0f - sg;                // d u_new / d v
        const float du_du = BETA * one_m;
        out_spk[idx]   = spk;
        out_u[idx]     = u_new;
        out_Eb[idx]    = fmaf(du_du, E_b[idx], one_m);
        ws_one_m[idx]  = one_m;
    }
}

// ---------------------------------------------------------------------------
// Kernel 2: E_W_new[b,i,j] = one_m[b,j] * (BETA*E_W[b,i,j] + x[b,i])
// Pure HBM-bound stream: 268 MB read + 268 MB write. float4, nontemporal.
// ---------------------------------------------------------------------------
__global__ __launch_bounds__(256) void ew_trace_update(
    const float* __restrict__ x,       // [64,1024]
    const float* __restrict__ E_W,     // [64,1024,1024]
    const float* __restrict__ one_m,   // [64,1024]
    float* __restrict__ out_EW)        // [64,1024,1024]
{
    const long long g  = (long long)blockIdx.x * blockDim.x + threadIdx.x; // float4 index
    const int jj = (int)(g & 255);          // j/4 within Dout
    const int i  = (int)((g >> 8) & 1023);  // Din index
    const int bs = (int)(g >> 18);          // batch sample

    const f32x4 om = ((const f32x4*)one_m)[bs * 256 + jj];  // hot, L2-resident
    const float xv = x[bs * 1024 + i];                      // broadcast scalar

    const f32x4 e = __builtin_nontemporal_load(((const f32x4*)E_W) + g);
    f32x4 res;
    res.x = om.x * fmaf(BETA, e.x, xv);
    res.y = om.y * fmaf(BETA, e.y, xv);
    res.z = om.z * fmaf(BETA, e.z, xv);
    res.w = om.w * fmaf(BETA, e.w, xv);
    __builtin_nontemporal_store(res, ((f32x4*)out_EW) + g);
}

extern "C" void kernel_launch(void* const* d_in, const int* in_sizes, int n_in,
                              void* d_out, int out_size, void* d_ws, size_t ws_size,
                              hipStream_t stream) {
    (void)in_sizes; (void)n_in; (void)out_size; (void)ws_size;
    const float* x   = (const float*)d_in[0];   // [64,1024]
    const float* W   = (const float*)d_in[1];   // [1024,1024]
    const float* b   = (const float*)d_in[2];   // [1024]
    const float* u0  = (const float*)d_in[3];   // [64,1024]
    const float* E_W = (const float*)d_in[4];   // [64,1024,1024]
    const float* E_b = (const float*)d_in[5];   // [64,1024]

    const int S       = 64 * 1024;            // 65536
    const long long EW = 64LL * 1024 * 1024;  // 67108864

    float* out_spk = (float*)d_out;           // return order: spk, u_new, E_W_new, E_b_new
    float* out_u   = out_spk + S;
    float* out_EW  = out_u + S;
    float* out_Eb  = out_EW + EW;

    float* ws_one_m = (float*)d_ws;           // 256 KB scratch

    // Kernel 1: 256 tiles = 32 blocks x 8 waves (wave32)
    dense_lif_wmma<<<32, 256, 0, stream>>>(x, W, b, u0, E_b,
                                           out_spk, out_u, out_Eb, ws_one_m);

    // Kernel 2: 16,777,216 float4s / 256 threads = 65536 blocks
    ew_trace_update<<<65536, 256, 0, stream>>>(x, E_W, ws_one_m, out_EW);
}